// StructureOnlyGNN_24481313587801
// MI455X (gfx1250) — compile-verified
//
#include <hip/hip_runtime.h>

typedef __attribute__((ext_vector_type(2))) float v2f;
typedef __attribute__((ext_vector_type(8))) float v8f;

#define IN_CH  128
#define HID_CH 128
#define OUT_CH 64

static __device__ __forceinline__ void atomAddF(float* p, float v) {
  __hip_atomic_fetch_add(p, v, __ATOMIC_RELAXED, __HIP_MEMORY_SCOPE_AGENT);
}

__global__ void fill_f32(float* __restrict__ p, float v, int n) {
  int i = blockIdx.x * blockDim.x + threadIdx.x;
  if (i < n) p[i] = v;
}

__global__ void deg_count(const long long* __restrict__ dst, float* __restrict__ deg, int E) {
  int e = blockIdx.x * blockDim.x + threadIdx.x;
  if (e < E) atomAddF(&deg[(int)dst[e]], 1.0f);
}

__global__ void rsqrt_inplace(float* __restrict__ p, int n) {
  int i = blockIdx.x * blockDim.x + threadIdx.x;
  if (i < n) p[i] = rsqrtf(p[i]);
}

// One wave computes one 16x16 output tile via chained V_WMMA_F32_16X16X4_F32.
// A: 16x4 f32 frag: lanes 0-15 hold row M=lane, K={k,k+1}; lanes 16-31 hold K={k+2,k+3}.
// B: 4x16 f32 frag: lanes 0-15 hold col N=lane for K rows {k,k+1}; lanes 16-31 for {k+2,k+3}.
// C/D: 8 VGPRs; lane half selects M base 0 / 8, col = lane&15.
template <int KDIM, int NDIM>
__global__ __launch_bounds__(128) void gemm_wmma_f32(
    const float* __restrict__ A, const float* __restrict__ B, float* __restrict__ C,
    int M, int totalTiles, int tilesN) {
  int wave = blockIdx.x * (blockDim.x >> 5) + (threadIdx.x >> 5);
  if (wave >= totalTiles) return;          // wave-uniform: EXEC stays all-1s for WMMA
  int tm = wave / tilesN;
  int tn = wave - tm * tilesN;
  int lane = threadIdx.x & 31;
  int lo = lane & 15;
  int hi = lane >> 4;                      // 0: K pair {0,1}, 1: K pair {2,3}

  int arow = tm * 16 + lo;
  if (arow >= M) arow = M - 1;             // safety clamp (N is a multiple of 16 here)
  int bcol = tn * 16 + lo;

  const float* ap = A + (size_t)arow * KDIM + (hi << 1);
  const float* bp = B + (size_t)(hi << 1) * NDIM + bcol;

  v8f acc = {};
#pragma unroll
  for (int k = 0; k < KDIM; k += 4) {
    v2f a = *(const v2f*)(ap + k);         // K=k+2h, k+2h+1 for this lane's row
    v2f b;
    b.x = bp[(size_t)k * NDIM];            // row K = k+2h
    b.y = bp[(size_t)(k + 1) * NDIM];      // row K = k+2h+1
    acc = __builtin_amdgcn_wmma_f32_16x16x4_f32(
        /*neg_a=*/false, a, /*neg_b=*/false, b,
        /*c_mod=*/(short)0, acc, /*reuse_a=*/false, /*reuse_b=*/false);
  }

  int rbase = tm * 16 + (hi << 3);
  float* out = C + (size_t)rbase * NDIM + bcol;
#pragma unroll
  for (int i = 0; i < 8; ++i)
    if (rbase + i < M) out[(size_t)i * NDIM] = acc[i];
}

// One wave per edge; 128 channels -> float4 per lane. Gather hits L2 (h fits in 192MB L2),
// accumulation via global_atomic_add_f32 in the L2 atomic units.
__global__ __launch_bounds__(256) void scatter128(
    const float* __restrict__ h, const long long* __restrict__ src,
    const long long* __restrict__ dst, const float* __restrict__ dinv,
    float* __restrict__ agg, int E) {
  int e = blockIdx.x * 8 + (threadIdx.x >> 5);
  if (e >= E) return;
  int lane = threadIdx.x & 31;
  int s = (int)src[e];
  int d = (int)dst[e];
  float coef = dinv[s] * dinv[d];
  float4 v = ((const float4*)(h + (size_t)s * 128))[lane];
  float* ap = agg + (size_t)d * 128 + lane * 4;
  atomAddF(ap + 0, v.x * coef);
  atomAddF(ap + 1, v.y * coef);
  atomAddF(ap + 2, v.z * coef);
  atomAddF(ap + 3, v.w * coef);
}

__global__ __launch_bounds__(256) void scatter64(
    const float* __restrict__ h, const long long* __restrict__ src,
    const long long* __restrict__ dst, const float* __restrict__ dinv,
    float* __restrict__ agg, int E) {
  int e = blockIdx.x * 8 + (threadIdx.x >> 5);
  if (e >= E) return;
  int lane = threadIdx.x & 31;
  int s = (int)src[e];
  int d = (int)dst[e];
  float coef = dinv[s] * dinv[d];
  float2 v = ((const float2*)(h + (size_t)s * 64))[lane];
  float* ap = agg + (size_t)d * 64 + lane * 2;
  atomAddF(ap + 0, v.x * coef);
  atomAddF(ap + 1, v.y * coef);
}

// agg = relu(agg + h*dinv^2 + b)  (in place; becomes layer-2 input)
__global__ void finish_relu128(float* __restrict__ agg, const float* __restrict__ h,
                               const float* __restrict__ dinv, const float* __restrict__ b,
                               int total) {
  int i = blockIdx.x * blockDim.x + threadIdx.x;
  if (i >= total) return;
  int node = i >> 7;
  int c = i & 127;
  float di = dinv[node];
  float v = agg[i] + h[i] * di * di + b[c];
  agg[i] = v > 0.0f ? v : 0.0f;
}

// out += h*dinv^2 + b  (out already holds neighbor aggregation)
__global__ void finish64(float* __restrict__ out, const float* __restrict__ h,
                         const float* __restrict__ dinv, const float* __restrict__ b,
                         int total) {
  int i = blockIdx.x * blockDim.x + threadIdx.x;
  if (i >= total) return;
  int node = i >> 6;
  int c = i & 63;
  float di = dinv[node];
  out[i] = out[i] + h[i] * di * di + b[c];
}

extern "C" void kernel_launch(void* const* d_in, const int* in_sizes, int n_in,
                              void* d_out, int out_size, void* d_ws, size_t ws_size,
                              hipStream_t stream) {
  const float* x      = (const float*)d_in[0];
  const long long* ei = (const long long*)d_in[1];   // int64 edge_index [2, E]
  const float* W1     = (const float*)d_in[2];
  const float* b1     = (const float*)d_in[3];
  const float* W2     = (const float*)d_in[4];
  const float* b2     = (const float*)d_in[5];
  float* out          = (float*)d_out;

  const int N = in_sizes[0] / IN_CH;
  const int E = in_sizes[1] / 2;
  const long long* src = ei;
  const long long* dst = ei + E;

  char* ws = (char*)d_ws;
  size_t off = 0;
  float* dinv = (float*)(ws + off); off += (((size_t)N * 4) + 255) & ~(size_t)255;
  float* h1   = (float*)(ws + off); off += (size_t)N * HID_CH * 4;
  float* agg1 = (float*)(ws + off); off += (size_t)N * HID_CH * 4;
  float* h2   = (float*)(ws + off); off += (size_t)N * OUT_CH * 4;
  (void)ws_size; (void)n_in; (void)out_size;

  const int T = 256;

  // --- degrees and normalization ---
  fill_f32<<<(N + T - 1) / T, T, 0, stream>>>(dinv, 1.0f, N);          // self-loop
  deg_count<<<(E + T - 1) / T, T, 0, stream>>>(dst, dinv, E);
  rsqrt_inplace<<<(N + T - 1) / T, T, 0, stream>>>(dinv, N);

  const int tilesM = (N + 15) / 16;

  // --- layer 1: h1 = x @ W1 ---
  {
    const int tilesN = HID_CH / 16;
    const int tot = tilesM * tilesN;
    gemm_wmma_f32<IN_CH, HID_CH><<<(tot + 3) / 4, 128, 0, stream>>>(x, W1, h1, N, tot, tilesN);
  }
  const int nh = N * HID_CH;
  fill_f32<<<(nh + T - 1) / T, T, 0, stream>>>(agg1, 0.0f, nh);
  scatter128<<<(E + 7) / 8, T, 0, stream>>>(h1, src, dst, dinv, agg1, E);
  finish_relu128<<<(nh + T - 1) / T, T, 0, stream>>>(agg1, h1, dinv, b1, nh);

  // --- layer 2: h2 = relu_out @ W2 ---
  {
    const int tilesN = OUT_CH / 16;
    const int tot = tilesM * tilesN;
    gemm_wmma_f32<HID_CH, OUT_CH><<<(tot + 3) / 4, 128, 0, stream>>>(agg1, W2, h2, N, tot, tilesN);
  }
  const int no = N * OUT_CH;
  fill_f32<<<(no + T - 1) / T, T, 0, stream>>>(out, 0.0f, no);
  scatter64<<<(E + 7) / 8, T, 0, stream>>>(h2, src, dst, dinv, out, E);
  finish64<<<(no + T - 1) / T, T, 0, stream>>>(out, h2, dinv, b2, no);
}